// LCAMatrixModel_41944650613376
// MI455X (gfx1250) — compile-verified
//
#include <hip/hip_runtime.h>
#include <hip/hip_bf16.h>

// ---------------------------------------------------------------------------
// o[i,j] = hard_sigmoid( sum_k selu(A[j,k] + B[i,k] + b1[k]) * W2[k] + b2 )
//   z = x @ W_enc + b_enc          [1024,128]   (WMMA f32 16x16x4)
//   A = z @ W1[:128] + b1          [1024,256]   (WMMA f32 16x16x4, b1 folded)
//   B = z @ W1[128:]               [1024,256]   (WMMA f32 16x16x4)
//   pairwise selu/dot phase        [1024,1024]  (async-to-LDS staging + VALU)
// ---------------------------------------------------------------------------

typedef __attribute__((ext_vector_type(2))) float v2f;
typedef __attribute__((ext_vector_type(8))) float v8f;

// ---------------------------------------------------------------------------
// Generic fp32 GEMM: C[M,N] = A[M,K] @ B[K,N] (+ bias[N]), one 16x16 tile per
// wave32 using V_WMMA_F32_16X16X4_F32.
//
// VGPR layouts per CDNA5 ISA 7.12.2:
//   A 16x4 : lane L -> M = L&15 ; VGPR0 = K=(2*hi), VGPR1 = K=(2*hi+1), hi=L>>4
//   B 4x16 : lane L -> N = L&15 ; VGPR0 = K=(2*hi), VGPR1 = K=(2*hi+1)
//   C/D    : VGPR r, lane L -> (M = r + 8*hi, N = L&15)
// ---------------------------------------------------------------------------
__global__ __launch_bounds__(256) void gemm_wmma_f32(
    const float* __restrict__ A, const float* __restrict__ B,
    const float* __restrict__ bias, float* __restrict__ C,
    int M, int N, int K, int lda, int ldb, int ldc)
{
    const int wavesPerBlock = blockDim.x >> 5;
    const int gw = blockIdx.x * wavesPerBlock + ((int)threadIdx.x >> 5);
    const int tilesN = N >> 4;
    const int totalTiles = (M >> 4) * tilesN;
    if (gw >= totalTiles) return;          // wave-uniform: EXEC stays all-ones

    const int mT = (gw / tilesN) << 4;
    const int nT = (gw % tilesN) << 4;

    const int lane = (int)threadIdx.x & 31;
    const int lo = lane & 15;
    const int hi = lane >> 4;              // 0 or 1
    const int kh = hi << 1;                // this lane-half holds K = kh, kh+1

    v8f acc = {};                          // zero accumulator

    const float* __restrict__ arow = A + (size_t)(mT + lo) * lda + kh;
    const float* __restrict__ bcol = B + nT + lo;

    for (int k0 = 0; k0 < K; k0 += 4) {
        v2f a, b;
        a.x = arow[k0];
        a.y = arow[k0 + 1];
        b.x = bcol[(size_t)(k0 + kh) * ldb];
        b.y = bcol[(size_t)(k0 + kh + 1) * ldb];
        // 8 args: (neg_a, A, neg_b, B, c_mod, C, reuse_a, reuse_b)
        acc = __builtin_amdgcn_wmma_f32_16x16x4_f32(
            false, a, false, b, (short)0, acc, false, false);
    }

    const int col = nT + lo;
    const float bv = bias ? bias[col] : 0.0f;
    const int rowbase = mT + (hi << 3);
#pragma unroll
    for (int r = 0; r < 8; ++r) {
        C[(size_t)(rowbase + r) * ldc + col] = acc[r] + bv;
    }
}

// ---------------------------------------------------------------------------
// Pairwise phase. 16x16 output tile per 256-thread block.
// Tile staging uses GLOBAL_LOAD_ASYNC_TO_LDS_B128 (CDNA5 async DMA path,
// tracked by ASYNCcnt) in GVS addressing mode:
//   global address is the 64-bit SGPR base plus the 32-bit VGPR offset; data lands at the LDS byte address in a VGPR
// LDS rows padded to 260 floats: 260 % 64 == 4, so the 16 distinct i-lanes
// land on 16 distinct banks (and shared rows broadcast) -> conflict-free.
// ---------------------------------------------------------------------------
#define HDIM 256
#define SROW 260

__device__ __forceinline__ void async_ld_b128(unsigned lds_byte, unsigned goff_byte,
                                              const void* base) {
    asm volatile("global_load_async_to_lds_b128 %0, %1, %2"
                 :: "v"(lds_byte), "v"(goff_byte), "s"(base)
                 : "memory");
}

__device__ __forceinline__ float selu_f(float x) {
    const float scale = 1.0507009873554805f;
    const float alpha = 1.6732632423543772f;
    float e = __expf(x);                   // v_exp_f32
    float neg = alpha * (e - 1.0f);
    return scale * (x > 0.0f ? x : neg);
}

__global__ __launch_bounds__(256) void pairwise_selu_dot(
    const float* __restrict__ Ap,   // [n, HDIM], b1 pre-folded
    const float* __restrict__ Bp,   // [n, HDIM]
    const float* __restrict__ W2,   // [HDIM]
    const float* __restrict__ b2,   // [1]
    float* __restrict__ out, int n)
{
    __shared__ float sA[16 * SROW];
    __shared__ float sB[16 * SROW];
    __shared__ float sW[HDIM];

    const int tid = (int)threadIdx.x;
    const int jT = (int)blockIdx.x << 4;   // column tile (A side)
    const int iT = (int)blockIdx.y << 4;   // row tile (B side)

    // --- Async stage: 16 rows x 256 floats (16 KB) each for A and B. -------
    {
        const int r  = tid >> 4;           // 0..15 row within tile
        const int c4 = tid & 15;           // float4 chunk lane within row
        // LDS byte addresses (the low 32 bits of a generic shared pointer are the LDS offset).
        const unsigned ldsA0 = (unsigned)(uintptr_t)(sA + r * SROW) + c4 * 16u;
        const unsigned ldsB0 = (unsigned)(uintptr_t)(sB + r * SROW) + c4 * 16u;
        // Global byte offsets relative to Ap/Bp.
        const unsigned gA0 = (unsigned)((jT + r) * HDIM + c4 * 4) * 4u;
        const unsigned gB0 = (unsigned)((iT + r) * HDIM + c4 * 4) * 4u;
#pragma unroll
        for (int t = 0; t < 4; ++t) {      // 4 chunks of 16 B per row-thread
            async_ld_b128(ldsA0 + t * 256u, gA0 + t * 256u, Ap);
            async_ld_b128(ldsB0 + t * 256u, gB0 + t * 256u, Bp);
        }
        sW[tid] = W2[tid];                 // overlaps with async DMA
    }
    asm volatile("s_wait_asynccnt 0x0" ::: "memory");  // this wave's DMAs done
    __syncthreads();                                    // whole tile visible

    const int il = tid & 15;               // i within tile
    const int jl = tid >> 4;               // j within tile
    const float* __restrict__ pa = sA + jl * SROW;
    const float* __restrict__ pb = sB + il * SROW;

    float acc = 0.0f;
#pragma unroll 4
    for (int k = 0; k < HDIM; k += 4) {
        const float4 av = *(const float4*)(pa + k);   // ds_load_b128 (broadcast)
        const float4 bv = *(const float4*)(pb + k);   // ds_load_b128 (16 banks)
        const float4 wv = *(const float4*)(sW + k);
        acc = fmaf(selu_f(av.x + bv.x), wv.x, acc);
        acc = fmaf(selu_f(av.y + bv.y), wv.y, acc);
        acc = fmaf(selu_f(av.z + bv.z), wv.z, acc);
        acc = fmaf(selu_f(av.w + bv.w), wv.w, acc);
    }

    float o = acc + b2[0];
    o = fminf(fmaxf(o + 3.0f, 0.0f), 6.0f) * (1.0f / 6.0f);   // hard_sigmoid
    out[(size_t)(iT + il) * n + (jT + jl)] = o;
}

// ---------------------------------------------------------------------------
extern "C" void kernel_launch(void* const* d_in, const int* in_sizes, int n_in,
                              void* d_out, int out_size, void* d_ws, size_t ws_size,
                              hipStream_t stream) {
    const float* x     = (const float*)d_in[0];   // [1024,128]
    const float* W_enc = (const float*)d_in[1];   // [128,128]
    const float* b_enc = (const float*)d_in[2];   // [128]
    const float* W1    = (const float*)d_in[3];   // [256,256]
    const float* b1    = (const float*)d_in[4];   // [256]
    const float* W2    = (const float*)d_in[5];   // [256]
    const float* b2    = (const float*)d_in[6];   // [1]
    float* out = (float*)d_out;

    const int N = 1024, RAW = 128, D = 128, H = 256;

    float* z  = (float*)d_ws;                 // 1024*128
    float* Ap = z + (size_t)N * D;            // 1024*256 (b1 folded in)
    float* Bp = Ap + (size_t)N * H;           // 1024*256
    // total workspace: 2.5 MiB

    // z = x @ W_enc + b_enc        (M=1024, N=128, K=128) -> 512 tiles / 8 waves
    gemm_wmma_f32<<<(N / 16) * (D / 16) / 8, 256, 0, stream>>>(
        x, W_enc, b_enc, z, N, D, RAW, RAW, D, D);

    // Ap = z @ W1[:128] + b1       (M=1024, N=256, K=128) -> 1024 tiles
    gemm_wmma_f32<<<(N / 16) * (H / 16) / 8, 256, 0, stream>>>(
        z, W1, b1, Ap, N, H, D, D, H, H);

    // Bp = z @ W1[128:]            (M=1024, N=256, K=128)
    gemm_wmma_f32<<<(N / 16) * (H / 16) / 8, 256, 0, stream>>>(
        z, W1 + (size_t)D * H, nullptr, Bp, N, H, D, D, H, H);

    // o[i,j] = hard_sigmoid(sum_k selu(Ap[j,k]+Bp[i,k]) * W2[k] + b2)
    dim3 grid(N / 16, N / 16);
    pairwise_selu_dot<<<grid, 256, 0, stream>>>(Ap, Bp, W2, b2, out, N);
}